// ECLayr_10728828306120
// MI455X (gfx1250) — compile-verified
//
#include <hip/hip_runtime.h>

#define STEPS 32
#define BATCH 32
#define PTS   256
#define HDIM  128
#define ODIM  64

#define CHUNKS 8        // blocks per batch for the edge-histogram pass
#define T1     256      // threads per histogram block

typedef __attribute__((ext_vector_type(2))) float v2f;
typedef __attribute__((ext_vector_type(8))) float v8f;

// -------------------------------------------------------------------------
// Kernel 1: per-batch edge-length histogram vs the 32-step filtration grid.
// hist[b][k] = #edges (i<j) with d(i,j) <= t_k   (exact integer counts)
// grid = BATCH*CHUNKS blocks x 256 threads; hist lives in d_ws (zeroed first).
// -------------------------------------------------------------------------
__global__ void ecc_hist_kernel(const float* __restrict__ x, int* __restrict__ hist) {
    const int b     = blockIdx.x / CHUNKS;
    const int chunk = blockIdx.x % CHUNKS;
    const int tid   = threadIdx.x;

    __shared__ float xs[PTS * 3];
    __shared__ float sq[PTS];
    __shared__ int   lhist[STEPS];

    for (int i = tid; i < PTS * 3; i += T1) xs[i] = x[b * PTS * 3 + i];
    if (tid < STEPS) lhist[tid] = 0;
    __syncthreads();
    for (int i = tid; i < PTS; i += T1) {
        float a0 = xs[i * 3 + 0], a1 = xs[i * 3 + 1], a2 = xs[i * 3 + 2];
        sq[i] = a0 * a0 + a1 * a1 + a2 * a2;
    }
    __syncthreads();

    int cnt[STEPS];
#pragma unroll
    for (int k = 0; k < STEPS; ++k) cnt[k] = 0;

    const int pairs_per_chunk = (PTS * PTS) / CHUNKS;   // 8192
    const int base = chunk * pairs_per_chunk;
    for (int p = tid; p < pairs_per_chunk; p += T1) {
        const int idx = base + p;
        const int i = idx & (PTS - 1);
        const int j = idx >> 8;
        float d = 3.0e38f;                 // invalid (i>=j) pairs never count
        if (i < j) {
            float dot = xs[i * 3 + 0] * xs[j * 3 + 0]
                      + xs[i * 3 + 1] * xs[j * 3 + 1]
                      + xs[i * 3 + 2] * xs[j * 3 + 2];
            float d2 = sq[i] + sq[j] - 2.0f * dot;      // matches reference form
            d = sqrtf(fmaxf(d2, 0.0f));
        }
#pragma unroll
        for (int k = 0; k < STEPS; ++k) {
            const float t = 2.0f * ((float)k / 31.0f);  // linspace(0,2,32), folded
            cnt[k] += (d <= t) ? 1 : 0;
        }
    }

#pragma unroll
    for (int k = 0; k < STEPS; ++k) atomicAdd(&lhist[k], cnt[k]);
    __syncthreads();
    if (tid < STEPS) atomicAdd(&hist[b * STEPS + tid], lhist[tid]);
}

// -------------------------------------------------------------------------
// Kernel 2: ecc = P - n_edges, then  out = relu(ecc@W1 + b1) @ W2 + b2
// Single block, 512 threads (16 waves). Full-precision V_WMMA_F32_16X16X4_F32.
// Stage 1: 2x8 tiles of 16x16 (one wave each), K=32 -> 8 wmma steps.
// Stage 2: 2x4 tiles (waves 0..7),              K=128 -> 32 wmma steps.
// -------------------------------------------------------------------------
#define ECC_LD 34    // padded row stride (banks: (2M+k) mod 64 distinct)
#define H_LD   130   // padded row stride for the hidden activations

__global__ void mlp_wmma_kernel(const int* __restrict__ hist,
                                const float* __restrict__ w1, const float* __restrict__ b1,
                                const float* __restrict__ w2, const float* __restrict__ b2,
                                float* __restrict__ out) {
    __shared__ float ecc[BATCH * ECC_LD];
    __shared__ float hsm[BATCH * H_LD];

    const int tid  = threadIdx.x;       // 0..511
    const int lane = tid & 31;
    const int wave = tid >> 5;          // 0..15
    const int half = lane >> 4;         // K-half select within the wave
    const int l16  = lane & 15;

    // ECC(t_k) = P - n_edges(t_k)
    for (int i = tid; i < BATCH * STEPS; i += 512) {
        const int m = i >> 5, k = i & 31;
        ecc[m * ECC_LD + k] = 256.0f - (float)hist[i];
    }
    __syncthreads();

    // ---------------- Stage 1: h = relu(ecc @ w1 + b1), [32 x 128] ----------
    {
        const int mt = wave >> 3;               // 0..1
        const int nt = wave & 7;                // 0..7
        const int M  = mt * 16 + l16;           // A-fragment row
        const int N  = nt * 16 + l16;           // B-fragment col
        v8f c = {};
#pragma unroll
        for (int kk = 0; kk < STEPS; kk += 4) {
            const int ka = kk + half * 2;       // lanes 16-31 carry K+2,K+3
            v2f a, bf;
            a.x  = ecc[M * ECC_LD + ka];
            a.y  = ecc[M * ECC_LD + ka + 1];
            bf.x = w1[(ka)     * HDIM + N];
            bf.y = w1[(ka + 1) * HDIM + N];
            c = __builtin_amdgcn_wmma_f32_16x16x4_f32(false, a, false, bf,
                                                      (short)0, c, false, false);
        }
        const float bias = b1[N];
#pragma unroll
        for (int r = 0; r < 8; ++r) {
            float v = c[r] + bias;
            v = v > 0.0f ? v : 0.0f;                        // ReLU
            const int row = mt * 16 + r + half * 8;         // C layout: M = r (+8)
            hsm[row * H_LD + N] = v;
        }
    }
    __syncthreads();

    // ---------------- Stage 2: out = h @ w2 + b2, [32 x 64] -----------------
    if (wave < 8) {                       // wave-uniform: EXEC all-ones inside
        const int mt = wave >> 2;         // 0..1
        const int nt = wave & 3;          // 0..3
        const int M  = mt * 16 + l16;
        const int N  = nt * 16 + l16;
        v8f c = {};
#pragma unroll
        for (int kk = 0; kk < HDIM; kk += 4) {
            const int ka = kk + half * 2;
            v2f a, bf;
            a.x  = hsm[M * H_LD + ka];
            a.y  = hsm[M * H_LD + ka + 1];
            bf.x = w2[(ka)     * ODIM + N];
            bf.y = w2[(ka + 1) * ODIM + N];
            c = __builtin_amdgcn_wmma_f32_16x16x4_f32(false, a, false, bf,
                                                      (short)0, c, false, false);
        }
        const float bias = b2[N];
#pragma unroll
        for (int r = 0; r < 8; ++r) {
            const int row = mt * 16 + r + half * 8;
            out[row * ODIM + N] = c[r] + bias;
        }
    }
}

// -------------------------------------------------------------------------
extern "C" void kernel_launch(void* const* d_in, const int* in_sizes, int n_in,
                              void* d_out, int out_size, void* d_ws, size_t ws_size,
                              hipStream_t stream) {
    const float* x  = (const float*)d_in[0];   // [32,256,3]
    const float* w1 = (const float*)d_in[1];   // [32,128]
    const float* b1 = (const float*)d_in[2];   // [128]
    const float* w2 = (const float*)d_in[3];   // [128,64]
    const float* b2 = (const float*)d_in[4];   // [64]
    float* out = (float*)d_out;                // [32,64]
    int* hist = (int*)d_ws;                    // [32][32] int

    hipMemsetAsync(hist, 0, BATCH * STEPS * sizeof(int), stream);
    ecc_hist_kernel<<<BATCH * CHUNKS, T1, 0, stream>>>(x, hist);
    mlp_wmma_kernel<<<1, 512, 0, stream>>>(hist, w1, b1, w2, b2, out);
}